// VMDK_24627342475570
// MI455X (gfx1250) — compile-verified
//
#include <hip/hip_runtime.h>
#include <math.h>

typedef __attribute__((ext_vector_type(2))) float v2f;
typedef __attribute__((ext_vector_type(8))) float v8f;

// y[m*rows + r] = dot(A[m][r][0..cols), x[m][0..cols)) + bias[m*rows + r]
// One wave32 computes a 16-row slab via V_WMMA_F32_16X16X4_F32, accumulating
// K in steps of 4. Only column N=0 of the B matrix carries the x vector.
// All waves in a block share the same matrix m, so x[m] is staged in LDS once
// and broadcast-read with ds_load_b128 (no divergence, no bank conflicts).
__global__ __launch_bounds__(256)
void gemv16_wmma(const float* __restrict__ A, const float* __restrict__ x,
                 const float* __restrict__ bias, float* __restrict__ y,
                 int rows, int cols, int tiles_per_mat, int total_tiles)
{
    __shared__ float xs[4096];                    // cols <= 4096 (16 KB of 320 KB LDS)

    const int wid   = threadIdx.x >> 5;
    const int lane  = threadIdx.x & 31;
    const int tile0 = blockIdx.x * (blockDim.x >> 5);
    const int m     = tile0 / tiles_per_mat;      // block-uniform component index

    // Cooperative stage of x[m] into LDS (regular temporal loads: x is hot/reused).
    for (int j = threadIdx.x; j < cols; j += blockDim.x)
        xs[j] = x[(size_t)m * cols + j];
    __syncthreads();

    const int tile = tile0 + wid;
    if (tile >= total_tiles) return;              // wave-uniform guard

    const int rt   = tile - m * tiles_per_mat;
    const int sub  = lane >> 4;                   // 0 or 1 (half-wave)
    const int row  = rt * 16 + (lane & 15);       // A-matrix row this lane feeds
    const int koff = sub << 1;                    // lanes 16-31 hold K=2,3

    const float* __restrict__ Arow = A + (size_t)m * rows * cols + (size_t)row * cols;
    const bool col0 = (lane & 15) == 0;           // lanes carrying B column N=0

    v8f c = {};
    for (int k0 = 0; k0 < cols; k0 += 4) {
        // A tile 16x4 f32: lane holds (row, k0+koff) and (row, k0+koff+1).
        // Weights are streamed exactly once -> non-temporal, keep L2 for x/v.
        v2f a = __builtin_nontemporal_load((const v2f*)(Arow + k0 + koff));

        // x[k0..k0+3] via one LDS b128 broadcast (same address in all lanes).
        float4 q = *(const float4*)(xs + k0);

        // B tile 4x16 f32: column 0 = x[k0..k0+3]. VGPR0 rows K=0(lo)/K=1(hi),
        // VGPR1 rows K=2(lo)/K=3(hi). Other columns zero -> register selects.
        float e0 = sub ? q.y : q.x;
        float e1 = sub ? q.w : q.z;
        v2f b;
        b.x = col0 ? e0 : 0.0f;
        b.y = col0 ? e1 : 0.0f;

        c = __builtin_amdgcn_wmma_f32_16x16x4_f32(
                /*neg_a=*/false, a, /*neg_b=*/false, b,
                /*c_mod=*/(short)0, c, /*reuse_a=*/false, /*reuse_b=*/false);
    }

    // D column N=0: lane 0 has rows rt*16+0..7 in c[0..7], lane 16 rows +8..15.
    if (col0) {
        const int rbase = rt * 16 + sub * 8;
        float* __restrict__       yo = y    + (size_t)m * rows + rbase;
        const float* __restrict__ bo = bias + (size_t)m * rows + rbase;
        #pragma unroll
        for (int g = 0; g < 8; ++g) yo[g] = c[g] + bo[g];
    }
}

// index == argmin(cumsum(nonneg)) == 0, so only codebook row 0 matters:
// sel[j] = (out[j] - K0[j]) * W[j] * W[j];  res = sigmoid(dot(sel, Wout) + bout)
__global__ __launch_bounds__(256)
void vmdk_finalize(const float* __restrict__ out, const float* __restrict__ K0,
                   const float* __restrict__ W, const float* __restrict__ Wout,
                   const float* __restrict__ bout, float* __restrict__ res, int D)
{
    __shared__ float red[256];
    float acc = 0.0f;
    for (int j = threadIdx.x; j < D; j += 256) {
        float df = (out[j] - K0[j]) * W[j];   // dis_feature[0][j]
        acc += df * W[j] * Wout[j];           // sel[j] * Wout[0][j]
    }
    red[threadIdx.x] = acc;
    __syncthreads();
    for (int s = 128; s > 0; s >>= 1) {
        if (threadIdx.x < s) red[threadIdx.x] += red[threadIdx.x + s];
        __syncthreads();
    }
    if (threadIdx.x == 0) {
        float z = red[0] + bout[0];
        res[0] = 1.0f / (1.0f + expf(-z));
    }
}

extern "C" void kernel_launch(void* const* d_in, const int* in_sizes, int n_in,
                              void* d_out, int out_size, void* d_ws, size_t ws_size,
                              hipStream_t stream) {
    (void)in_sizes; (void)n_in; (void)out_size; (void)ws_size;
    const float* x    = (const float*)d_in[0];   // [8,1,1024]
    // d_in[1..4] (Wq,bq,Wk,bk) are dead: softmax over S=1 key is identically 1.
    const float* Wv   = (const float*)d_in[5];   // [8,4096,1024]
    const float* bv   = (const float*)d_in[6];   // [8,4096]
    const float* Wo   = (const float*)d_in[7];   // [8,1024,4096]
    const float* bo   = (const float*)d_in[8];   // [8,1024]
    const float* K    = (const float*)d_in[9];   // [4096,8192]; only row 0 reachable
    const float* W    = (const float*)d_in[10];  // [8192]
    const float* Wout = (const float*)d_in[11];  // [1,8192]
    const float* bout = (const float*)d_in[12];  // [1]

    float* v    = (float*)d_ws;                  // 8*4096 floats
    float* outv = v + 8 * 4096;                  // 8*1024 floats

    // Stage 1: v[m] = Wv[m] @ x[m] + bv[m]. 8*4096 rows -> 2048 wave-tiles -> 256 blocks.
    gemv16_wmma<<<256, 256, 0, stream>>>(Wv, x, bv, v, 4096, 1024, 256, 2048);
    // Stage 2: out[m] = Wo[m] @ v[m] + bo[m]. 8*1024 rows -> 512 wave-tiles -> 64 blocks.
    gemv16_wmma<<<64, 256, 0, stream>>>(Wo, v, bo, outv, 1024, 4096, 64, 512);
    // Stage 3: codebook row 0 distance -> linear -> sigmoid.
    vmdk_finalize<<<1, 256, 0, stream>>>(outv, K, W, Wout, bout, (float*)d_out, 8192);
}